// SAGEDepthEmb_80676665688557
// MI455X (gfx1250) — compile-verified
//
#include <hip/hip_runtime.h>
#include <hip/hip_bf16.h>

#define N_NODES 50000
#define F_HID   512
#define F_OUT   47
#define N_EDGES 800000
#define BN_EPS  1e-5f

typedef __attribute__((ext_vector_type(2))) float v2f;
typedef __attribute__((ext_vector_type(8))) float v8f;

// ---------------------------------------------------------------------------
// Degree count: cnt[dst[e]] += 1
// ---------------------------------------------------------------------------
__global__ void degree_kernel(const int* __restrict__ dst, float* __restrict__ cnt, int E) {
    int i = blockIdx.x * blockDim.x + threadIdx.x;
    if (i < E) unsafeAtomicAdd(&cnt[dst[i]], 1.0f);
}

__global__ void invcnt_kernel(const float* __restrict__ cnt, float* __restrict__ invc, int n) {
    int i = blockIdx.x * blockDim.x + threadIdx.x;
    if (i < n) invc[i] = 1.0f / fmaxf(cnt[i], 1.0f);
}

// ---------------------------------------------------------------------------
// Edge scatter-add: agg[dst[e], :] += h[src[e], :]   (F fixed at 512)
// One float4 per (edge, 4-feature-chunk); 128 consecutive threads share an edge
// so source reads are fully coalesced and atomics hit consecutive addresses.
// ---------------------------------------------------------------------------
__global__ void scatter_kernel(const float* __restrict__ h, const int* __restrict__ src,
                               const int* __restrict__ dst, float* __restrict__ agg, int E) {
    const long long total  = (long long)E * 128;      // 512/4 chunks per edge
    const long long stride = (long long)gridDim.x * blockDim.x;
    for (long long idx = (long long)blockIdx.x * blockDim.x + threadIdx.x;
         idx < total; idx += stride) {
        const int e  = (int)(idx >> 7);
        const int f4 = (int)(idx & 127);
        const int s  = src[e];
        const int d  = dst[e];
        const float4 v = ((const float4*)(h + (size_t)s * F_HID))[f4];
        float* o = agg + (size_t)d * F_HID + f4 * 4;
        unsafeAtomicAdd(o + 0, v.x);
        unsafeAtomicAdd(o + 1, v.y);
        unsafeAtomicAdd(o + 2, v.z);
        unsafeAtomicAdd(o + 3, v.w);
    }
}

// ---------------------------------------------------------------------------
// SAGE GEMM:  Out[i,j] = sum_k (agg[i,k]*invc[i]) * Wl[k,j]
//                      + sum_k  X[i,k]            * Wr[k,j]  + bias[j]
// 256 threads / block, 64x32 output tile, 8 waves each owning 16x16 via
// V_WMMA_F32_16X16X4_F32 accumulation. K staged in LDS in 32-wide chunks;
// the two source matrices are folded into one 2*K chunk loop.
// ---------------------------------------------------------------------------
#define TM 64
#define TN 32
#define KC 32

__global__ __launch_bounds__(256)
void sage_gemm_kernel(const float* __restrict__ Agg, const float* __restrict__ invc,
                      const float* __restrict__ X,
                      const float* __restrict__ Wl, const float* __restrict__ Wr,
                      const float* __restrict__ bias, float* __restrict__ Out,
                      int nrows, int K, int nout) {
    __shared__ float As[TM][KC + 4];
    __shared__ float Bs[KC][TN + 4];

    const int tid  = threadIdx.x;
    const int lane = tid & 31;
    const int wave = tid >> 5;
    const int mi   = wave & 3;   // 0..3 -> M sub-tile
    const int ni   = wave >> 2;  // 0..1 -> N sub-tile
    const int rowBase = blockIdx.x * TM;
    const int colBase = blockIdx.y * TN;

    v8f acc = {};

    const int chunksPerMat = K / KC;       // 16
    const int nChunks      = 2 * chunksPerMat;

    for (int kc = 0; kc < nChunks; ++kc) {
        const bool firstMat = (kc < chunksPerMat);
        const int  k0 = (firstMat ? kc : kc - chunksPerMat) * KC;
        const float* __restrict__ A = firstMat ? Agg : X;
        const float* __restrict__ B = firstMat ? Wl  : Wr;

        __syncthreads();   // protect previous chunk's LDS reads

        // ---- stage A: 64 rows x 32 K-floats (2 passes of 32 rows) ----
        {
            const int r = tid >> 3;            // 0..31
            const int c = (tid & 7) << 2;      // 0,4,...,28
            #pragma unroll
            for (int p = 0; p < 2; ++p) {
                const int row  = r + p * 32;
                const int grow = rowBase + row;
                float4 v = make_float4(0.f, 0.f, 0.f, 0.f);
                if (grow < nrows) {
                    v = *(const float4*)(A + (size_t)grow * K + k0 + c);
                    if (firstMat) {
                        const float s = invc[grow];
                        v.x *= s; v.y *= s; v.z *= s; v.w *= s;
                    }
                }
                As[row][c + 0] = v.x; As[row][c + 1] = v.y;
                As[row][c + 2] = v.z; As[row][c + 3] = v.w;
            }
        }
        // ---- stage B: 32 K-rows x 32 cols (single pass) ----
        {
            const int r  = tid >> 3;           // K row 0..31
            const int c  = (tid & 7) << 2;     // col chunk
            const int gc = colBase + c;
            const float* bp = B + (size_t)(k0 + r) * nout + gc;
            float4 v = make_float4(0.f, 0.f, 0.f, 0.f);
            if (gc + 3 < nout) {
                v = *(const float4*)bp;
            } else {
                if (gc + 0 < nout) v.x = bp[0];
                if (gc + 1 < nout) v.y = bp[1];
                if (gc + 2 < nout) v.z = bp[2];
                if (gc + 3 < nout) v.w = bp[3];
            }
            Bs[r][c + 0] = v.x; Bs[r][c + 1] = v.y;
            Bs[r][c + 2] = v.z; Bs[r][c + 3] = v.w;
        }
        __syncthreads();

        // ---- WMMA accumulate over this K chunk ----
        const int am   = mi * 16 + (lane & 15);        // A row (M)
        const int bn   = ni * 16 + (lane & 15);        // B col (N)
        const int koff = (lane >> 4) << 1;             // 0 (lanes 0-15) / 2 (16-31)
        #pragma unroll
        for (int kk = 0; kk < KC; kk += 4) {
            v2f a, b;
            a.x = As[am][kk + koff];
            a.y = As[am][kk + koff + 1];
            b.x = Bs[kk + koff][bn];
            b.y = Bs[kk + koff + 1][bn];
            acc = __builtin_amdgcn_wmma_f32_16x16x4_f32(
                /*neg_a=*/false, a, /*neg_b=*/false, b,
                /*c_mod=*/(short)0, acc, /*reuse_a=*/false, /*reuse_b=*/false);
        }
    }

    // ---- epilogue: +bias, store 16x16 tile per wave ----
    const int gcol = colBase + ni * 16 + (lane & 15);
    if (gcol < nout) {
        const float bv = bias[gcol];
        #pragma unroll
        for (int r = 0; r < 8; ++r) {
            const int grow = rowBase + mi * 16 + r + ((lane >> 4) << 3);
            if (grow < nrows)
                Out[(size_t)grow * nout + gcol] = acc[r] + bv;
        }
    }
}

// ---------------------------------------------------------------------------
// BatchNorm stats: per-column sum / sumsq over rows (F fixed at 512)
// blockDim.x == 512 (one thread per column), grid-stride over rows.
// ---------------------------------------------------------------------------
__global__ void bn_stats_kernel(const float* __restrict__ h, float* __restrict__ sums,
                                int nrows) {
    const int col = threadIdx.x;
    float s = 0.f, sq = 0.f;
    for (int r = blockIdx.x; r < nrows; r += gridDim.x) {
        const float v = h[(size_t)r * F_HID + col];
        s  += v;
        sq += v * v;
    }
    unsafeAtomicAdd(&sums[col], s);
    unsafeAtomicAdd(&sums[F_HID + col], sq);
}

__global__ void bn_apply_relu_kernel(float* __restrict__ h, const float* __restrict__ sums,
                                     const float* __restrict__ gamma,
                                     const float* __restrict__ beta, int nrows) {
    const long long total  = (long long)nrows * F_HID;
    const long long stride = (long long)gridDim.x * blockDim.x;
    const float inv_n = 1.0f / (float)nrows;
    for (long long idx = (long long)blockIdx.x * blockDim.x + threadIdx.x;
         idx < total; idx += stride) {
        const int col = (int)(idx & (F_HID - 1));
        const float mu  = sums[col] * inv_n;
        const float var = sums[F_HID + col] * inv_n - mu * mu;
        const float v = (h[idx] - mu) * rsqrtf(var + BN_EPS) * gamma[col] + beta[col];
        h[idx] = fmaxf(v, 0.f);
    }
}

// ---------------------------------------------------------------------------
// Launch
// ---------------------------------------------------------------------------
extern "C" void kernel_launch(void* const* d_in, const int* in_sizes, int n_in,
                              void* d_out, int out_size, void* d_ws, size_t ws_size,
                              hipStream_t stream) {
    const float* x    = (const float*)d_in[0];
    const int*   edge = (const int*)d_in[1];
    const int*   src  = edge;             // edge_index[0]
    const int*   dst  = edge + N_EDGES;   // edge_index[1]
    const float* Wl0 = (const float*)d_in[2];
    const float* Wr0 = (const float*)d_in[3];
    const float* b0  = (const float*)d_in[4];
    const float* g0  = (const float*)d_in[5];
    const float* be0 = (const float*)d_in[6];
    const float* Wl1 = (const float*)d_in[7];
    const float* Wr1 = (const float*)d_in[8];
    const float* b1  = (const float*)d_in[9];
    const float* g1  = (const float*)d_in[10];
    const float* be1 = (const float*)d_in[11];
    const float* Wl2 = (const float*)d_in[12];
    const float* Wr2 = (const float*)d_in[13];
    const float* b2  = (const float*)d_in[14];
    float* out = (float*)d_out;

    // workspace layout
    const size_t featElems = (size_t)N_NODES * F_HID;
    float* agg  = (float*)d_ws;
    float* h1   = agg  + featElems;
    float* h2   = h1   + featElems;
    float* cnt  = h2   + featElems;
    float* invc = cnt  + N_NODES;
    float* sums = invc + N_NODES;

    // degrees (same for every layer)
    hipMemsetAsync(cnt, 0, N_NODES * sizeof(float), stream);
    degree_kernel<<<(N_EDGES + 255) / 256, 256, 0, stream>>>(dst, cnt, N_EDGES);
    invcnt_kernel<<<(N_NODES + 255) / 256, 256, 0, stream>>>(cnt, invc, N_NODES);

    const dim3 gemmBlock(256);
    const dim3 gemmGridHid((N_NODES + TM - 1) / TM, (F_HID + TN - 1) / TN);
    const dim3 gemmGridOut((N_NODES + TM - 1) / TM, (F_OUT + TN - 1) / TN);

    auto sage_layer = [&](const float* hin, const float* Wl, const float* Wr,
                          const float* b, float* hout, int nout, dim3 grid) {
        hipMemsetAsync(agg, 0, featElems * sizeof(float), stream);
        scatter_kernel<<<8192, 256, 0, stream>>>(hin, src, dst, agg, N_EDGES);
        sage_gemm_kernel<<<grid, gemmBlock, 0, stream>>>(
            agg, invc, hin, Wl, Wr, b, hout, N_NODES, F_HID, nout);
    };
    auto bn_relu = [&](float* h, const float* g, const float* be) {
        hipMemsetAsync(sums, 0, 2 * F_HID * sizeof(float), stream);
        bn_stats_kernel<<<240, F_HID, 0, stream>>>(h, sums, N_NODES);
        bn_apply_relu_kernel<<<4096, 256, 0, stream>>>(h, sums, g, be, N_NODES);
    };

    // layer 0
    sage_layer(x, Wl0, Wr0, b0, h1, F_HID, gemmGridHid);
    bn_relu(h1, g0, be0);
    // layer 1
    sage_layer(h1, Wl1, Wr1, b1, h2, F_HID, gemmGridHid);
    bn_relu(h2, g1, be1);
    // output layer
    sage_layer(h2, Wl2, Wr2, b2, out, F_OUT, gemmGridOut);
}